// Relation_Trans_36833639531232
// MI455X (gfx1250) — compile-verified
//
#include <hip/hip_runtime.h>

// ---------------------------------------------------------------------------
// Relation_Trans forward for MI455X (gfx1250), fp32 WMMA (v_wmma_f32_16x16x4_f32)
//
//   person   : [2048,4096] @ [4096,256] -> ReLU -> @ [256,256]  => [2048,256]
//   pairs    : 4032 ordered (i,j), i!=j per batch, concat(pf_i, pf_j) -> 512
//   relation : [129024,512] @ [512,256] -> ReLU -> @ [256,256]  => [129024,256]
//
// Fused two-layer tiles (hidden in LDS), pair tensor never materialized,
// global<->LDS staging via CDNA5 async-to-LDS ops (ASYNCcnt) when available.
// ---------------------------------------------------------------------------

typedef float v2f __attribute__((ext_vector_type(2)));
typedef float v4f __attribute__((ext_vector_type(4)));
typedef float v8f __attribute__((ext_vector_type(8)));
typedef int   v4i __attribute__((ext_vector_type(4)));

#define GAS __attribute__((address_space(1)))   // global
#define LAS __attribute__((address_space(3)))   // LDS

#define SA 20    // A-tile stride: 20r mod 64 distinct for r<16; 80B rows (16B align)
#define SH 260   // hidden/out stride: 260r = 4r mod 64 distinct; 1040B rows (16B align)

#if defined(__HIP_DEVICE_COMPILE__) && __has_builtin(__builtin_amdgcn_global_load_async_to_lds_b128)
#define USE_ASYNC_LD 1
#else
#define USE_ASYNC_LD 0
#endif
#if defined(__HIP_DEVICE_COMPILE__) && __has_builtin(__builtin_amdgcn_global_store_async_from_lds_b128)
#define USE_ASYNC_ST 1
#else
#define USE_ASYNC_ST 0
#endif

#if USE_ASYNC_LD || USE_ASYNC_ST
#if __has_builtin(__builtin_amdgcn_s_wait_asynccnt)
#define WAIT_ASYNC() __builtin_amdgcn_s_wait_asynccnt(0)
#else
#define WAIT_ASYNC() asm volatile("s_wait_asynccnt 0x0" ::: "memory")
#endif
#else
#define WAIT_ASYNC()
#endif

#if USE_ASYNC_LD
__device__ __forceinline__ void g2l_b128(const float* g, float* l) {
    __builtin_amdgcn_global_load_async_to_lds_b128((GAS v4i*)g, (LAS v4i*)l, 0, 0);
}
#endif
#if USE_ASYNC_ST
__device__ __forceinline__ void l2g_b128(float* g, const float* l) {
    __builtin_amdgcn_global_store_async_from_lds_b128((GAS v4i*)g, (LAS v4i*)l, 0, 0);
}
#endif

// One workgroup = 256 threads = 8 wave32, computes 32 rows x 256 cols.
// Wave w owns col-blocks {2w,2w+1} x row-blocks {0,1} -> four 16x16 D tiles.
template <int K1, bool GATHER>
__global__ __launch_bounds__(256) void mlp2_wmma_kernel(
    const float* __restrict__ Asrc,   // GATHER ? person_feature[2048,256] : x[2048,4096]
    const float* __restrict__ W1,     // [K1, 256] row-major (K-major)
    const float* __restrict__ b1,     // [256]
    const float* __restrict__ W2,     // [256, 256]
    const float* __restrict__ b2,     // [256]
    float* __restrict__ out)          // [rows, 256]
{
    __shared__ __align__(16) float As[32 * SA];    //  2560 B : A chunk 32 x 16
    __shared__ __align__(16) float Bs[16 * 256];   // 16384 B : W chunk 16 x 256
    __shared__ __align__(16) float Hs[32 * SH];    // 33280 B : hidden / out tile
    __shared__ unsigned baseI[32];
    __shared__ unsigned baseJ[32];

    const int tid  = threadIdx.x;
    const int lane = tid & 31;
    const int wave = tid >> 5;
    const int hf   = lane >> 4;           // half-wave: K pair select / row+8
    const int m    = lane & 15;           // col-in-tile (B/D) or row-in-tile (A)
    const int cb0  = wave * 2;
    const int row0 = (int)blockIdx.x * 32;

    if (GATHER) {
        if (tid < 32) {
            unsigned p   = (unsigned)(row0 + tid);
            unsigned bb  = p / 4032u;
            unsigned rem = p % 4032u;
            unsigned i   = rem / 63u;
            unsigned q   = rem % 63u;
            unsigned j   = q + (q >= i ? 1u : 0u);
            baseI[tid] = (bb * 64u + i) * 256u;
            baseJ[tid] = (bb * 64u + j) * 256u;
        }
        __syncthreads();
    }

    // A-staging coordinates: threads 0..127 move one float4 each of the 32x16 chunk
    const int sr  = tid >> 2;             // A row 0..31
    const int sk4 = (tid & 3) * 4;        // A col 0,4,8,12

    // ---- Layer 1 accumulators, bias-splat init (a lane's 8 elems share a column)
    v8f acc[2][2];
    {
        float bias[2] = { b1[cb0 * 16 + m], b1[(cb0 + 1) * 16 + m] };
#pragma unroll
        for (int rb = 0; rb < 2; ++rb)
#pragma unroll
            for (int cbi = 0; cbi < 2; ++cbi)
#pragma unroll
                for (int g = 0; g < 8; ++g)
                    acc[rb][cbi][g] = bias[cbi];
    }

    // ---- Layer 1: C = A[32,K1] * W1[K1,256], K chunked by 16
    for (int kb = 0; kb < K1; kb += 16) {
        // stage A chunk (32x16) and W chunk (16x256) into LDS
        const float* asrc = nullptr;
        if (tid < 128) {
            if (GATHER) {
                unsigned base = (kb < 256) ? (baseI[sr] + kb) : (baseJ[sr] + (kb - 256));
                asrc = Asrc + base + sk4;
            } else {
                asrc = Asrc + (size_t)(row0 + sr) * K1 + kb + sk4;
            }
        }
#if USE_ASYNC_LD
        if (tid < 128) g2l_b128(asrc, &As[sr * SA + sk4]);
#pragma unroll
        for (int i = 0; i < 4; ++i) {
            int e = tid + i * 256;
            int k = e >> 6, c4 = (e & 63) * 4;
            g2l_b128(&W1[(size_t)(kb + k) * 256 + c4], &Bs[k * 256 + c4]);
        }
        WAIT_ASYNC();
#else
        v4f t[4], ta;
        if (tid < 128) ta = *(const v4f*)asrc;
#pragma unroll
        for (int i = 0; i < 4; ++i) {
            int e = tid + i * 256;
            int k = e >> 6, c4 = (e & 63) * 4;
            t[i] = *(const v4f*)&W1[(size_t)(kb + k) * 256 + c4];
        }
        if (tid < 128) *(v4f*)&As[sr * SA + sk4] = ta;
#pragma unroll
        for (int i = 0; i < 4; ++i) {
            int e = tid + i * 256;
            int k = e >> 6, c4 = (e & 63) * 4;
            *(v4f*)&Bs[k * 256 + c4] = t[i];
        }
#endif
        __syncthreads();

#pragma unroll
        for (int kk = 0; kk < 16; kk += 4) {
            v2f a[2], b[2];
#pragma unroll
            for (int rb = 0; rb < 2; ++rb) {
                a[rb].x = As[(rb * 16 + m) * SA + kk + 2 * hf + 0];
                a[rb].y = As[(rb * 16 + m) * SA + kk + 2 * hf + 1];
            }
#pragma unroll
            for (int cbi = 0; cbi < 2; ++cbi) {
                int n = (cb0 + cbi) * 16 + m;
                b[cbi].x = Bs[(kk + 2 * hf + 0) * 256 + n];
                b[cbi].y = Bs[(kk + 2 * hf + 1) * 256 + n];
            }
#pragma unroll
            for (int rb = 0; rb < 2; ++rb)
#pragma unroll
                for (int cbi = 0; cbi < 2; ++cbi)
                    acc[rb][cbi] = __builtin_amdgcn_wmma_f32_16x16x4_f32(
                        false, a[rb], false, b[cbi], (short)0, acc[rb][cbi],
                        false, false);
        }
        __syncthreads();
    }

    // ---- ReLU, park hidden tile in LDS
#pragma unroll
    for (int rb = 0; rb < 2; ++rb)
#pragma unroll
        for (int cbi = 0; cbi < 2; ++cbi) {
            int col = (cb0 + cbi) * 16 + m;
#pragma unroll
            for (int g = 0; g < 8; ++g) {
                int row = rb * 16 + g + 8 * hf;
                float v = acc[rb][cbi][g];
                Hs[row * SH + col] = v > 0.0f ? v : 0.0f;
            }
        }
    __syncthreads();

    // ---- Layer 2 accumulators with bias
    {
        float bias[2] = { b2[cb0 * 16 + m], b2[(cb0 + 1) * 16 + m] };
#pragma unroll
        for (int rb = 0; rb < 2; ++rb)
#pragma unroll
            for (int cbi = 0; cbi < 2; ++cbi)
#pragma unroll
                for (int g = 0; g < 8; ++g)
                    acc[rb][cbi][g] = bias[cbi];
    }

    // ---- Layer 2: D = ReLU(h)[32,256] * W2[256,256]
    for (int kb = 0; kb < 256; kb += 16) {
#if USE_ASYNC_LD
#pragma unroll
        for (int i = 0; i < 4; ++i) {
            int e = tid + i * 256;
            int k = e >> 6, c4 = (e & 63) * 4;
            g2l_b128(&W2[(size_t)(kb + k) * 256 + c4], &Bs[k * 256 + c4]);
        }
        WAIT_ASYNC();
#else
        v4f t[4];
#pragma unroll
        for (int i = 0; i < 4; ++i) {
            int e = tid + i * 256;
            int k = e >> 6, c4 = (e & 63) * 4;
            t[i] = *(const v4f*)&W2[(size_t)(kb + k) * 256 + c4];
        }
#pragma unroll
        for (int i = 0; i < 4; ++i) {
            int e = tid + i * 256;
            int k = e >> 6, c4 = (e & 63) * 4;
            *(v4f*)&Bs[k * 256 + c4] = t[i];
        }
#endif
        __syncthreads();

#pragma unroll
        for (int kk = 0; kk < 16; kk += 4) {
            v2f a[2], b[2];
#pragma unroll
            for (int rb = 0; rb < 2; ++rb) {
                a[rb].x = Hs[(rb * 16 + m) * SH + kb + kk + 2 * hf + 0];
                a[rb].y = Hs[(rb * 16 + m) * SH + kb + kk + 2 * hf + 1];
            }
#pragma unroll
            for (int cbi = 0; cbi < 2; ++cbi) {
                int n = (cb0 + cbi) * 16 + m;
                b[cbi].x = Bs[(kk + 2 * hf + 0) * 256 + n];
                b[cbi].y = Bs[(kk + 2 * hf + 1) * 256 + n];
            }
#pragma unroll
            for (int rb = 0; rb < 2; ++rb)
#pragma unroll
                for (int cbi = 0; cbi < 2; ++cbi)
                    acc[rb][cbi] = __builtin_amdgcn_wmma_f32_16x16x4_f32(
                        false, a[rb], false, b[cbi], (short)0, acc[rb][cbi],
                        false, false);
        }
        __syncthreads();
    }

    // ---- Park D tile in LDS (reuse Hs), then fully-coalesced b128 stores
#pragma unroll
    for (int rb = 0; rb < 2; ++rb)
#pragma unroll
        for (int cbi = 0; cbi < 2; ++cbi) {
            int col = (cb0 + cbi) * 16 + m;
#pragma unroll
            for (int g = 0; g < 8; ++g) {
                int row = rb * 16 + g + 8 * hf;
                Hs[row * SH + col] = acc[rb][cbi][g];
            }
        }
    __syncthreads();

#pragma unroll                              // 32x256 floats = 2048 float4, 8 per thread
    for (int e = tid; e < 2048; e += 256) {
        int r = e >> 6, c4 = (e & 63) * 4;
#if USE_ASYNC_ST
        l2g_b128(&out[(size_t)(row0 + r) * 256 + c4], &Hs[r * SH + c4]);
#else
        *(v4f*)&out[(size_t)(row0 + r) * 256 + c4] = *(const v4f*)&Hs[r * SH + c4];
#endif
    }
    WAIT_ASYNC();   // drain async stores before wave end
}

extern "C" void kernel_launch(void* const* d_in, const int* in_sizes, int n_in,
                              void* d_out, int out_size, void* d_ws, size_t ws_size,
                              hipStream_t stream) {
    const float* x    = (const float*)d_in[0];  // [32,64,16,256] = [2048,4096]
    const float* W1_t = (const float*)d_in[1];  // [4096,256]
    const float* b1_t = (const float*)d_in[2];  // [256]
    const float* W2_t = (const float*)d_in[3];  // [256,256]
    const float* b2_t = (const float*)d_in[4];  // [256]
    const float* W1_i = (const float*)d_in[5];  // [512,256]
    const float* b1_i = (const float*)d_in[6];  // [256]
    const float* W2_i = (const float*)d_in[7];  // [256,256]
    const float* b2_i = (const float*)d_in[8];  // [256]

    const size_t REL_ELEMS = (size_t)32 * 4032 * 256;    // 33,030,144
    const size_t PF_ELEMS  = (size_t)32 * 64 * 16 * 256; // 8,388,608

    float* rel   = (float*)d_out;                 // relation_feature
    float* pf    = rel + REL_ELEMS;               // pf == x (stop_gradient identity)
    float* pfeat = pf + PF_ELEMS;                 // person_feature [2048,256]

    // pf output = x (forward identity)
    (void)hipMemcpyAsync(pf, x, PF_ELEMS * sizeof(float), hipMemcpyDeviceToDevice, stream);

    // person MLP: 2048 rows -> 64 workgroups
    mlp2_wmma_kernel<4096, false><<<64, 256, 0, stream>>>(
        x, W1_t, b1_t, W2_t, b2_t, pfeat);

    // interaction MLP: 129024 pair rows -> 4032 workgroups, gathers pf_i|pf_j
    mlp2_wmma_kernel<512, true><<<4032, 256, 0, stream>>>(
        pfeat, W1_i, b1_i, W2_i, b2_i, rel);
}